// DynamicModel_3487513444947
// MI455X (gfx1250) — compile-verified
//
#include <hip/hip_runtime.h>
#include <hip/hip_bf16.h>

typedef __attribute__((ext_vector_type(2))) float v2f;
typedef __attribute__((ext_vector_type(8))) float v8f;

#define TP 10
#define JP 31
#define NCOMBO 75            // NTAU*NE*NX = 3*5*5
static __device__ __constant__ float kLOG2PI = 1.8378770664093453f;

// One wave (32 threads) per individual i.
// Utility GEMM: U[75x31] = F[75x4] @ G[4x31] via v_wmma_f32_16x16x4_f32,
// tiled 5 row-tiles x 2 col-tiles.
__global__ __launch_bounds__(32) void mig_like_kernel(
    const float* __restrict__ wages,        // [N,T]
    const float* __restrict__ skill,        // [N]
    const float* __restrict__ dist,         // [J,J]
    const float* __restrict__ amenity,      // [J]
    const float* __restrict__ region_wage,  // [J]
    const float* __restrict__ prem,         // [J]
    const float* __restrict__ alpha_tau,    // [3]
    const float* __restrict__ beta_skill,   // [1]
    const float* __restrict__ gamma_,       // [1]
    const float* __restrict__ pi_tau_raw,   // [3]
    const float* __restrict__ eta_grid,     // [5]
    const float* __restrict__ nu_grid,      // [5]
    const float* __restrict__ xi_grid,      // [5]
    const float* __restrict__ sigma_raw,    // [5]
    const float* __restrict__ p_e_raw,      // [5]
    const float* __restrict__ p_n_raw,      // [5]
    const float* __restrict__ p_x_raw,      // [5]
    const float* __restrict__ p_s_raw,      // [5]
    const int*   __restrict__ j_obs,        // [N,T]
    const int*   __restrict__ loc_prev,     // [N,T]
    const int*   __restrict__ home_idx,     // [N]
    float* __restrict__ per_i)              // [N] out: log(L_i + 1e-10)
{
  const int i    = blockIdx.x;
  const int lane = threadIdx.x;
  const int h    = lane >> 4;   // lane half: selects K={0,1} vs K={2,3}
  const int l    = lane & 15;   // M (A) / N (B,D) index within tile

  const float skill_i = skill[i];
  const float gam     = gamma_[0];
  const int   home_i  = home_idx[i];

  // ---- A tiles: F[c] = (alpha_tau, eta*skill, xi, 1); K = 2h + reg ----
  float a0[5], a1[5];
#pragma unroll
  for (int rt = 0; rt < 5; ++rt) {
    int c = rt * 16 + l;
    float f0 = 0.f, f1 = 0.f, f2 = 0.f, f3 = 0.f;
    if (c < NCOMBO) {
      int tau = c / 25, rem = c % 25, e = rem / 5, x = rem % 5;
      f0 = alpha_tau[tau];
      f1 = eta_grid[e] * skill_i;
      f2 = xi_grid[x];
      f3 = 1.0f;
    }
    a0[rt] = (h == 0) ? f0 : f2;   // VGPR0: K=0 (h=0) / K=2 (h=1)
    a1[rt] = (h == 0) ? f1 : f3;   // VGPR1: K=1 (h=0) / K=3 (h=1)
  }

  // ---- static B parts: G0=amenity, G1=skill*prem, G2=home indicator ----
  float b0s[2], b1stat[2];
#pragma unroll
  for (int jt = 0; jt < 2; ++jt) {
    int j = jt * 16 + l;
    float am = 0.f, sp = 0.f, hm = 0.f;
    if (j < JP) {
      am = amenity[j];
      sp = skill_i * prem[j];
      hm = (j == home_i) ? 1.0f : 0.0f;
    }
    b0s[jt]   = (h == 0) ? am : hm;   // VGPR0: K=0 / K=2
    b1stat[jt]= sp;                    // h==0 VGPR1 (K=1); h==1 is per-t dist
  }

  float acc[5][8];                     // log_Lam accumulators (replicated over l)
#pragma unroll
  for (int rt = 0; rt < 5; ++rt)
#pragma unroll
    for (int r = 0; r < 8; ++r) acc[rt][r] = 0.f;

  const v8f czero = {};

  for (int t = 0; t < TP; ++t) {
    const int jp = loc_prev[i * TP + t];
    const int jo = j_obs  [i * TP + t];

    // dynamic B part: G3[j] = -gamma * dist[jp, j]  (K=3 -> h==1, VGPR1)
    float mc0 = -gam * dist[jp * JP + l];
    float mc1 = ((16 + l) < JP) ? (-gam * dist[jp * JP + 16 + l]) : 0.f;
    v2f B0, B1;
    B0[0] = b0s[0]; B0[1] = (h == 0) ? b1stat[0] : mc0;
    B1[0] = b0s[1]; B1[1] = (h == 0) ? b1stat[1] : mc1;

    const int jo_hi = jo >> 4;
    const int srcl  = (lane & 16) | (jo & 15);

#pragma unroll
    for (int rt = 0; rt < 5; ++rt) {
      v2f Av; Av[0] = a0[rt]; Av[1] = a1[rt];
      v8f d0 = __builtin_amdgcn_wmma_f32_16x16x4_f32(
          false, Av, false, B0, (short)0, czero, false, false);
      v8f d1 = __builtin_amdgcn_wmma_f32_16x16x4_f32(
          false, Av, false, B1, (short)0, czero, false, false);
#pragma unroll
      for (int r = 0; r < 8; ++r) {
        float u0 = d0[r], u1 = d1[r];
        // per-lane partial sum of exp over both column tiles (mask j=31)
        float pe = __expf(u0) + ((l == 15) ? 0.f : __expf(u1));
        pe += __shfl_xor(pe, 1, 16);
        pe += __shfl_xor(pe, 2, 16);
        pe += __shfl_xor(pe, 4, 16);
        pe += __shfl_xor(pe, 8, 16);
        // utility at observed destination (from matching tile / lane)
        float usel = jo_hi ? u1 : u0;
        float uo = __shfl(usel, srcl, 32);
        int c = rt * 16 + 8 * h + r;      // combo this (r, half) holds
        float lam = uo - __logf(pe);
        acc[rt][r] += (c < NCOMBO) ? lam : 0.f;
      }
    }
  }

  // ---- wage density: lane k<25 handles (n,s); e unrolled ----
  const float beta = beta_skill[0];
  float dvals[TP];
#pragma unroll
  for (int t = 0; t < TP; ++t) {
    int jo = j_obs[i * TP + t];
    dvals[t] = wages[i * TP + t] - (region_wage[jo] + beta * skill_i);
  }

  float sn = 0.f, ss_ = 0.f, sx = 0.f, se = 0.f, st = 0.f;
#pragma unroll
  for (int k = 0; k < 5; ++k) { sn += p_n_raw[k]; ss_ += p_s_raw[k]; sx += p_x_raw[k]; se += p_e_raw[k]; }
#pragma unroll
  for (int k = 0; k < 3; ++k) st += pi_tau_raw[k];

  const bool act = lane < 25;
  const int  nidx = act ? (lane / 5) : 0;
  const int  sidx = act ? (lane % 5) : 0;
  const float sg   = sigma_raw[sidx] + 0.5f;
  const float w_ns = (p_n_raw[nidx] / sn) * (p_s_raw[sidx] / ss_);
  const float inv2s2 = 0.5f / (sg * sg);
  const float base = -((float)TP) * (__logf(sg) + 0.5f * kLOG2PI);
  const float nuv  = nu_grid[nidx];

  float Ae[5];
#pragma unroll
  for (int e = 0; e < 5; ++e) {
    float m = eta_grid[e] + nuv;
    float ssum = 0.f;
#pragma unroll
    for (int t = 0; t < TP; ++t) { float z = dvals[t] - m; ssum += z * z; }
    float Psi = base - ssum * inv2s2;
    float pa = act ? (w_ns * __expf(Psi)) : 0.f;
    pa += __shfl_xor(pa, 16, 32);
    pa += __shfl_xor(pa, 8, 32);
    pa += __shfl_xor(pa, 4, 32);
    pa += __shfl_xor(pa, 2, 32);
    pa += __shfl_xor(pa, 1, 32);
    Ae[e] = pa;                         // all lanes hold A[e]
  }

  // ---- combine: B[tau,e] = sum_x p_x exp(Lam); fully unrolled ----
  float pxv[5], pev[5], ptv[3];
#pragma unroll
  for (int k = 0; k < 5; ++k) { pxv[k] = p_x_raw[k] / sx; pev[k] = p_e_raw[k] / se; }
#pragma unroll
  for (int k = 0; k < 3; ++k) ptv[k] = pi_tau_raw[k] / st;

  float Bacc[3][5];
#pragma unroll
  for (int tt = 0; tt < 3; ++tt)
#pragma unroll
    for (int e = 0; e < 5; ++e) Bacc[tt][e] = 0.f;

#pragma unroll
  for (int rt = 0; rt < 5; ++rt) {
#pragma unroll
    for (int r = 0; r < 8; ++r) {
      float vlo = __shfl(acc[rt][r], 0, 32);   // combo c = rt*16 + r
      float vhi = __shfl(acc[rt][r], 16, 32);  // combo c = rt*16 + 8 + r
      const int c1 = rt * 16 + r;
      const int c2 = c1 + 8;
      if (c1 < NCOMBO) {
        const int tau = c1 / 25, rem = c1 % 25, e = rem / 5, x = rem % 5;
        Bacc[tau][e] += pxv[x] * __expf(vlo);
      }
      if (c2 < NCOMBO) {
        const int tau = c2 / 25, rem = c2 % 25, e = rem / 5, x = rem % 5;
        Bacc[tau][e] += pxv[x] * __expf(vhi);
      }
    }
  }

  float L = 0.f;
#pragma unroll
  for (int tau = 0; tau < 3; ++tau) {
    float lt = 0.f;
#pragma unroll
    for (int e = 0; e < 5; ++e) lt += pev[e] * Ae[e] * Bacc[tau][e];
    L += ptv[tau] * lt;
  }

  if (lane == 0) per_i[i] = __logf(L + 1e-10f);
}

// Deterministic fixed-order reduction of per-individual logL into scalar.
__global__ __launch_bounds__(256) void reduce_sum_kernel(
    const float* __restrict__ per_i, float* __restrict__ out, int n)
{
  __shared__ float sm[256];
  float s = 0.f;
  for (int idx = threadIdx.x; idx < n; idx += 256) s += per_i[idx];
  sm[threadIdx.x] = s;
  __syncthreads();
  for (int off = 128; off > 0; off >>= 1) {
    if (threadIdx.x < off) sm[threadIdx.x] += sm[threadIdx.x + off];
    __syncthreads();
  }
  if (threadIdx.x == 0) out[0] = sm[0];
}

extern "C" void kernel_launch(void* const* d_in, const int* in_sizes, int n_in,
                              void* d_out, int out_size, void* d_ws, size_t ws_size,
                              hipStream_t stream) {
  const float* wages       = (const float*)d_in[0];
  const float* skill       = (const float*)d_in[1];
  const float* dist        = (const float*)d_in[2];
  const float* amenity     = (const float*)d_in[3];
  const float* region_wage = (const float*)d_in[4];
  const float* prem        = (const float*)d_in[5];
  const float* alpha_tau   = (const float*)d_in[6];
  const float* beta_skill  = (const float*)d_in[7];
  const float* gamma_      = (const float*)d_in[8];
  const float* pi_tau_raw  = (const float*)d_in[9];
  const float* eta_grid    = (const float*)d_in[10];
  const float* nu_grid     = (const float*)d_in[11];
  const float* xi_grid     = (const float*)d_in[12];
  const float* sigma_raw   = (const float*)d_in[13];
  const float* p_e_raw     = (const float*)d_in[14];
  const float* p_n_raw     = (const float*)d_in[15];
  const float* p_x_raw     = (const float*)d_in[16];
  const float* p_s_raw     = (const float*)d_in[17];
  const int*   j_obs       = (const int*)d_in[18];
  const int*   loc_prev    = (const int*)d_in[19];
  const int*   home_idx    = (const int*)d_in[20];

  const int n = in_sizes[1];            // N individuals (skill is [N])
  float* per_i = (float*)d_ws;

  mig_like_kernel<<<n, 32, 0, stream>>>(
      wages, skill, dist, amenity, region_wage, prem,
      alpha_tau, beta_skill, gamma_, pi_tau_raw,
      eta_grid, nu_grid, xi_grid, sigma_raw,
      p_e_raw, p_n_raw, p_x_raw, p_s_raw,
      j_obs, loc_prev, home_idx, per_i);

  reduce_sum_kernel<<<1, 256, 0, stream>>>(per_i, (float*)d_out, n);
}